// ExtremeFMLayer_77558519431754
// MI455X (gfx1250) — compile-verified
//
#include <hip/hip_runtime.h>

// ExtremeFM bilinear layer on gfx1250 via V_WMMA_F32_16X16X4_F32.
//
// out[b,l,d] = sum_{i,j} x0[b,i,d] * x1[b,j,d] * W[i*64+j, l]
//            = sum_i x0[b,i,d] * ( W2_i[16(l) x 64(j)] @ X1_b[64(j) x 16(d)] )[l,d]
//
// W2 (1024x64 f32, 256KB) lives in LDS in native WMMA A-fragment order.
// Each wave handles 2 batches (2 independent accumulator chains sharing the
// A fragment), B fragments (X1_b) are register-resident across all 64 i-blocks.

typedef __attribute__((ext_vector_type(2))) float v2f;
typedef __attribute__((ext_vector_type(8))) float v8f;

#define F1N 64
#define F2N 64
#define DN  16
#define LN  16

extern "C" __global__ __launch_bounds__(256)
void fm_wmma_kernel(const float* __restrict__ x0,
                    const float* __restrict__ x1,
                    const float* __restrict__ filt,
                    float* __restrict__ out, int B) {
    // 64 i-blocks * 16 k-steps * 32 lanes * 2 floats = 65536 f32 = 256 KB
    extern __shared__ float w2frag[];

    const int tid = threadIdx.x;

    // ---- Stage 1: gather filters -> WMMA A-fragment layout in LDS ----
    // Fragment (i,k): A[l, jj] = W2[(i*16+l), 4k+jj] = filt[(i*64 + 4k+jj)*16 + l]
    // ISA A-layout (16x4 f32): lane<16 holds M=lane, K={0,1}; lane>=16 holds M=lane-16, K={2,3}
    for (int s = tid; s < F1N * 16 * 32; s += 256) {
        const int lane = s & 31;
        const int k    = (s >> 5) & 15;
        const int i    = s >> 9;
        const int l    = lane & 15;
        const int jj   = (lane >> 4) * 2;
        const int j    = 4 * k + jj;
        v2f av;
        av.x = filt[(i * F2N + j)     * LN + l];
        av.y = filt[(i * F2N + j + 1) * LN + l];
        *reinterpret_cast<v2f*>(&w2frag[2 * s]) = av;
    }
    __syncthreads();

    const int lane = tid & 31;
    const int wave = blockIdx.x * (blockDim.x >> 5) + (tid >> 5);
    const int d    = lane & 15;
    const int hi   = lane >> 4;           // 0: lanes 0-15, 1: lanes 16-31
    const int b0   = wave * 2;
    const int b1   = b0 + 1;
    if (b0 >= B) return;                  // wave-uniform: EXEC stays all-1s

    // ---- Stage 2: preload B fragments (X1_b columns) into registers ----
    // ISA B-layout (4x16 f32): vgpr v, half h -> K = v + 2h, N = lane&15
    v2f bfa[16], bfb[16];
    const float* x1a = x1 + (size_t)b0 * (F2N * DN);
    const float* x1b = x1 + (size_t)b1 * (F2N * DN);
#pragma unroll
    for (int k = 0; k < 16; ++k) {
        const int j = 4 * k + 2 * hi;
        bfa[k].x = x1a[j * DN + d];
        bfa[k].y = x1a[(j + 1) * DN + d];
        bfb[k].x = x1b[j * DN + d];
        bfb[k].y = x1b[(j + 1) * DN + d];
    }

    const float* x0a = x0 + (size_t)b0 * (F1N * DN) + d;
    const float* x0b = x0 + (size_t)b1 * (F1N * DN) + d;

    v8f oa = {};   // out accumulators: vgpr v, half h -> l = v + 8h, column d
    v8f ob = {};

    // ---- Stage 3: 64 i-blocks; each = K=64 WMMA chain + rank-1 epilogue ----
    for (int i = 0; i < F1N; ++i) {
        const float s0 = x0a[i * DN];     // x0[b0, i, d]  (per-lane scalar)
        const float s1 = x0b[i * DN];
        v8f c0 = {};
        v8f c1 = {};
        const float* wbase = &w2frag[i * 16 * 64];
#pragma unroll
        for (int k = 0; k < 16; ++k) {
            v2f a = *reinterpret_cast<const v2f*>(wbase + k * 64 + lane * 2);
            c0 = __builtin_amdgcn_wmma_f32_16x16x4_f32(
                     false, a, false, bfa[k], (short)0, c0, false, false);
            c1 = __builtin_amdgcn_wmma_f32_16x16x4_f32(
                     false, a, false, bfb[k], (short)0, c1, false, false);
        }
#pragma unroll
        for (int v = 0; v < 8; ++v) {
            oa[v] += s0 * c0[v];
            ob[v] += s1 * c1[v];
        }
    }

    // ---- Stage 4: write out[b, l, d] ----
    float* outa = out + (size_t)b0 * (LN * DN);
    float* outb = out + (size_t)b1 * (LN * DN);
#pragma unroll
    for (int v = 0; v < 8; ++v) {
        outa[(v + 8 * hi) * DN + d] = oa[v];
        outb[(v + 8 * hi) * DN + d] = ob[v];
    }
}

extern "C" void kernel_launch(void* const* d_in, const int* in_sizes, int n_in,
                              void* d_out, int out_size, void* d_ws, size_t ws_size,
                              hipStream_t stream) {
    (void)n_in; (void)out_size; (void)d_ws; (void)ws_size;
    const float* x0   = (const float*)d_in[0];   // [B, 64, 16]
    const float* x1   = (const float*)d_in[1];   // [B, 64, 16]
    const float* filt = (const float*)d_in[2];   // [1, 4096, 16]
    float* out = (float*)d_out;                  // [B, 16, 16]

    const int B = in_sizes[0] / (F1N * DN);      // 2048
    const int blocks = (B + 15) / 16;            // 16 batches per 256-thread block
    const size_t shmem = 65536 * sizeof(float);  // 256 KB (within 320 KB WGP LDS)

    (void)hipFuncSetAttribute((const void*)fm_wmma_kernel,
                              hipFuncAttributeMaxDynamicSharedMemorySize,
                              (int)shmem);
    fm_wmma_kernel<<<blocks, 256, shmem, stream>>>(x0, x1, filt, out, B);
}